// Attention_42348377538911
// MI455X (gfx1250) — compile-verified
//
#include <hip/hip_runtime.h>
#include <hip/hip_bf16.h>

// Problem constants (from reference): B=2, C=128, N=16^3=4096, H=4, Dh=32.
#define NB    2
#define CIN   128
#define NSP   4096
#define HEADS 4
#define DHEAD 32
#define OQKV  384   // 3 * HEADS * DHEAD
#define QSCALE 0.17677669529663687f  // 1/sqrt(32)

typedef float v2f __attribute__((ext_vector_type(2)));
typedef float v8f __attribute__((ext_vector_type(8)));
typedef int   v4i __attribute__((vector_size(16)));    // pointee type the async builtin wants

typedef __attribute__((address_space(1))) v4i* glb_v4i_p;   // global (device) ptr
typedef __attribute__((address_space(3))) v4i* lds_v4i_p;   // LDS ptr

__device__ __forceinline__ v8f wmma_f32(v2f a, v2f b, v8f c) {
  // V_WMMA_F32_16X16X4_F32 : D = A(16x4) * B(4x16) + C(16x16), fp32 throughout
  return __builtin_amdgcn_wmma_f32_16x16x4_f32(
      /*neg_a=*/false, a, /*neg_b=*/false, b,
      /*c_mod=*/(short)0, c, /*reuse_a=*/false, /*reuse_b=*/false);
}

// --- gfx1250 async copy global->LDS (ASYNCcnt-tracked DMA), 16B per lane ---
__device__ __forceinline__ void async_copy_b128(const float* gsrc, float* ldst) {
#if __has_builtin(__builtin_amdgcn_global_load_async_to_lds_b128)
  __builtin_amdgcn_global_load_async_to_lds_b128(
      (glb_v4i_p)(void*)gsrc,
      (lds_v4i_p)(void*)ldst,
      /*offset=*/0, /*cpol=*/0);
#else
  unsigned lds_off = (unsigned)(unsigned long)(__attribute__((address_space(3))) float*)ldst;
  asm volatile("global_load_async_to_lds_b128 %0, %1, off"
               :: "v"(lds_off), "v"(gsrc) : "memory");
#endif
}

__device__ __forceinline__ void wait_async0() {
#if __has_builtin(__builtin_amdgcn_s_wait_asynccnt)
  __builtin_amdgcn_s_wait_asynccnt(0);
#else
  asm volatile("s_wait_asynccnt 0x0" ::: "memory");
#endif
}

// One wave computes a 16x16 f32 C tile of  C = A(row-major MxK) * B(row-major KxN).
// f32 16x16x4 operand layouts:
//   A: lane holds row m = lane&15; VGPR v holds K = 2*(lane>>4)+v  (contiguous -> float2)
//   B: lane holds col n = lane&15; VGPR v holds K = 2*(lane>>4)+v  (stride-ldb gathers)
//   C: reg r holds row m = r + 8*(lane>>4), col n = lane&15
__device__ __forceinline__ v8f tile_gemm_f32(const float* __restrict__ A, int lda,
                                             const float* __restrict__ Bm, int ldb,
                                             int m0, int n0, int K) {
  const int lane = threadIdx.x & 31;
  const int col  = lane & 15;
  const int hi   = lane >> 4;
  v8f acc = {0.f, 0.f, 0.f, 0.f, 0.f, 0.f, 0.f, 0.f};
  const float* arow = A + (size_t)(m0 + col) * lda + 2 * hi;      // + k + v contiguous
  const float* bcol = Bm + (size_t)(2 * hi) * ldb + n0 + col;     // + (k+v)*ldb
  for (int k = 0; k < K; k += 4) {
    v2f a = *(const v2f*)(arow + k);            // 8B aligned: lda,k,2*hi all even
    v2f b;
    b.x = bcol[(size_t)k * ldb];
    b.y = bcol[(size_t)(k + 1) * ldb];
    acc = wmma_f32(a, b, acc);
  }
  return acc;
}

// ---------------- Kernel 1: qkv = w_qkv @ x, q-rows pre-scaled ----------------
__global__ __launch_bounds__(256) void qkv_proj_kernel(const float* __restrict__ x,
                                                       const float* __restrict__ w_qkv,
                                                       float* __restrict__ qkv) {
  const int wave = threadIdx.x >> 5;
  const int tile = blockIdx.x * 8 + wave;              // 2 * 24 * 256 = 12288 tiles
  const int b  = tile / (24 * 256);
  const int r  = tile % (24 * 256);
  const int mo = (r / 256) * 16;                       // output-channel tile (0..383)
  const int no = (r % 256) * 16;                       // spatial tile
  const float* xb = x + (size_t)b * CIN * NSP;
  float* qb = qkv + (size_t)b * OQKV * NSP;

  v8f acc = tile_gemm_f32(w_qkv, CIN, xb, NSP, mo, no, CIN);
  const float mul = (mo < HEADS * DHEAD) ? QSCALE : 1.0f;  // scale only Q rows

  const int lane = threadIdx.x & 31;
  const int col  = lane & 15;
  const int hi   = lane >> 4;
#pragma unroll
  for (int q = 0; q < 8; ++q)
    qb[(size_t)(mo + q + 8 * hi) * NSP + no + col] = acc[q] * mul;
}

// ---------------- Kernel 2: flash attention ----------------
// Block = one (b,h) + 8 consecutive 16-row i-tiles (one per wave).
// Per j-step the whole block stages the shared 32x16 K and V tiles into LDS with
// one async b128 copy per thread (4KB total), cutting K/V L2 traffic 8x.
__global__ __launch_bounds__(256) void flash_attn_kernel(const float* __restrict__ qkv,
                                                         float* __restrict__ mid) {
  __shared__ __align__(16) float kvbuf[1024];          // K tile [32][16] | V tile [32][16]
  __shared__ __align__(16) float plds[8 * 256];        // per-wave P tiles
  const int tid  = threadIdx.x;
  const int wave = tid >> 5;
  const int lane = tid & 31;
  const int col  = lane & 15;
  const int hi   = lane >> 4;
  float* pbuf = plds + wave * 256;
  float* kbuf = kvbuf;                                  // [d][16]
  float* vbuf = kvbuf + 512;                            // [d][16]

  const int blk = blockIdx.x;                           // 2*4*32 = 256 blocks exactly
  const int b   = blk / (HEADS * 32);
  const int rr  = blk % (HEADS * 32);
  const int h   = rr / 32;
  const int i0  = (rr % 32) * 128 + wave * 16;          // this wave's 16 query rows

  const float* qb = qkv + ((size_t)b * OQKV + h * DHEAD) * NSP;           // [32][N]
  const float* kb = qkv + ((size_t)b * OQKV + 128 + h * DHEAD) * NSP;     // [32][N]
  const float* vb = qkv + ((size_t)b * OQKV + 256 + h * DHEAD) * NSP;     // [32][N]

  // Staging plan: tid<128 copies K, tid>=128 copies V; 4 floats (16B) per thread.
  const int e = (tid & 127) * 4;                        // element idx in 32x16 tile
  const float* gsrc = ((tid < 128) ? kb : vb) + (size_t)(e >> 4) * NSP + (e & 15);
  float* ldst = ((tid < 128) ? kbuf : vbuf) + e;

  // Hoisted Q fragments (loop-invariant over j): A(i_local, d) = q[d][i0+i_local]
  const float* qp = qb + (size_t)(2 * hi) * NSP + i0 + col;
  v2f qf[8];
#pragma unroll
  for (int t = 0; t < 8; ++t) {
    qf[t].x = qp[(size_t)(4 * t) * NSP];
    qf[t].y = qp[(size_t)(4 * t + 1) * NSP];
  }

  v8f o0 = {0.f,0.f,0.f,0.f,0.f,0.f,0.f,0.f};           // out cols d = 0..15
  v8f o1 = {0.f,0.f,0.f,0.f,0.f,0.f,0.f,0.f};           // out cols d = 16..31
  float m_r[8], l_r[8];
#pragma unroll
  for (int q = 0; q < 8; ++q) { m_r[q] = -1e30f; l_r[q] = 0.f; }

  for (int j0 = 0; j0 < NSP; j0 += 16) {
    // ---- async-stage K/V j-tile into LDS (ASYNCcnt DMA path) ----
    async_copy_b128(gsrc + j0, ldst);
    wait_async0();
    __syncthreads();                                    // all copies visible

    // ---- S tile = Q^T K  (16 i x 16 j, K = d = 32 -> 8 WMMAs) ----
    v8f s = {0.f,0.f,0.f,0.f,0.f,0.f,0.f,0.f};
#pragma unroll
    for (int dk = 0; dk < DHEAD; dk += 4) {
      v2f bv;                                           // B(d, j) from LDS K tile
      bv.x = kbuf[(dk + 2 * hi) * 16 + col];
      bv.y = kbuf[(dk + 2 * hi + 1) * 16 + col];
      s = wmma_f32(qf[dk >> 2], bv, s);
    }

    // ---- online softmax; rows of S live across 16 lanes of each half ----
#pragma unroll
    for (int q = 0; q < 8; ++q) {
      float sv = s[q];
      float mx = sv;
      mx = fmaxf(mx, __shfl_xor(mx, 1));
      mx = fmaxf(mx, __shfl_xor(mx, 2));
      mx = fmaxf(mx, __shfl_xor(mx, 4));
      mx = fmaxf(mx, __shfl_xor(mx, 8));
      const float mnew = fmaxf(m_r[q], mx);
      const float corr = __expf(m_r[q] - mnew);
      const float p    = __expf(sv - mnew);
      float ps = p;
      ps += __shfl_xor(ps, 1);
      ps += __shfl_xor(ps, 2);
      ps += __shfl_xor(ps, 4);
      ps += __shfl_xor(ps, 8);
      l_r[q] = l_r[q] * corr + ps;
      m_r[q] = mnew;
      o0[q] *= corr;
      o1[q] *= corr;
      pbuf[(q + 8 * hi) * 16 + col] = p;                // C-layout row -> LDS row-major
    }
    __syncthreads();                                    // P tiles ready

    // ---- O += P * V^T : M=i(16), N=d(2 x 16), K=j(16) -> 8 WMMAs ----
#pragma unroll
    for (int js = 0; js < 16; js += 4) {
      v2f a  = *(const v2f*)(pbuf + col * 16 + js + 2 * hi);        // A(i, j)
      v2f b0 = *(const v2f*)(vbuf + col * 16 + js + 2 * hi);        // B(j, d), d=col
      v2f b1 = *(const v2f*)(vbuf + (16 + col) * 16 + js + 2 * hi); // d=16+col
      o0 = wmma_f32(a, b0, o0);
      o1 = wmma_f32(a, b1, o1);
    }
    __syncthreads();                                    // K/V tile consumed; safe to restage
  }

  // ---- normalize and write mid[b][h*32 + d][i] ----
  float* mb = mid + ((size_t)b * CIN + h * DHEAD) * NSP;
#pragma unroll
  for (int q = 0; q < 8; ++q) {
    const float inv = 1.0f / l_r[q];
    const int   i   = i0 + q + 8 * hi;
    mb[(size_t)col * NSP + i]        = o0[q] * inv;
    mb[(size_t)(16 + col) * NSP + i] = o1[q] * inv;
  }
}

// ---------------- Kernel 3: y = w_out @ mid + b_out ----------------
__global__ __launch_bounds__(256) void out_proj_kernel(const float* __restrict__ mid,
                                                       const float* __restrict__ w_out,
                                                       const float* __restrict__ b_out,
                                                       float* __restrict__ y) {
  const int wave = threadIdx.x >> 5;
  const int tile = blockIdx.x * 8 + wave;              // 2 * 8 * 256 = 4096 tiles
  const int b  = tile / (8 * 256);
  const int r  = tile % (8 * 256);
  const int mo = (r / 256) * 16;
  const int no = (r % 256) * 16;
  const float* mb = mid + (size_t)b * CIN * NSP;
  float* yb = y + (size_t)b * CIN * NSP;

  v8f acc = tile_gemm_f32(w_out, CIN, mb, NSP, mo, no, CIN);

  const int lane = threadIdx.x & 31;
  const int col  = lane & 15;
  const int hi   = lane >> 4;
#pragma unroll
  for (int q = 0; q < 8; ++q) {
    const int row = mo + q + 8 * hi;
    yb[(size_t)row * NSP + no + col] = acc[q] + b_out[row];
  }
}

extern "C" void kernel_launch(void* const* d_in, const int* in_sizes, int n_in,
                              void* d_out, int out_size, void* d_ws, size_t ws_size,
                              hipStream_t stream) {
  const float* x     = (const float*)d_in[0];   // [2,128,16,16,16]
  const float* w_qkv = (const float*)d_in[1];   // [384,128]
  const float* w_out = (const float*)d_in[2];   // [128,128]
  const float* b_out = (const float*)d_in[3];   // [128]
  float* y = (float*)d_out;                     // [2,128,16,16,16]

  float* qkv = (float*)d_ws;                               // 2*384*4096 floats (12 MB)
  float* mid = qkv + (size_t)NB * OQKV * NSP;              // 2*128*4096 floats (4 MB)

  // 12288 tiles / 8 waves per block
  qkv_proj_kernel<<<1536, 256, 0, stream>>>(x, w_qkv, qkv);
  // 256 blocks: one (b,h) + 128 query rows each
  flash_attn_kernel<<<256, 256, 0, stream>>>(qkv, mid);
  // 4096 tiles / 8 waves per block
  out_proj_kernel<<<512, 256, 0, stream>>>(mid, w_out, b_out, y);
}